// PostProcessor_60292750901577
// MI455X (gfx1250) — compile-verified
//
#include <hip/hip_runtime.h>

// ---------------- problem constants (from the reference) ----------------
#define NBOX      216320            // 16 * 52 * 52 * 5 boxes
#define VALS      85                // 5 + 80 values per box (contiguous in preds)
#define BPB       32                // boxes decoded per block (32*85*4 = 10880 B LDS)
#define THREADS_DEC 256             // 8 waves -> 4 boxes per wave
#define CHUNKS    ((BPB * VALS * 4) / 16)   // 680 x 16-byte async copies, exact
#define NEG_F     (-1.0e9f)
#define GRID_WH   52.0f

// ---------------- CDNA5 async global->LDS path (inline asm) ----------------
__device__ __forceinline__ void async_copy16(unsigned lds_off, const void* gaddr) {
    // GLOBAL_LOAD_ASYNC_TO_LDS_B128: vdst = LDS byte address, vaddr = 64-bit global addr
    asm volatile("global_load_async_to_lds_b128 %0, %1, off"
                 :: "v"(lds_off), "v"(gaddr)
                 : "memory");
}
__device__ __forceinline__ void async_wait0() {
    asm volatile("s_wait_asynccnt 0x0" ::: "memory");
}

// ---------------- kernel 1: decode + softmax + compaction ----------------
__global__ __launch_bounds__(THREADS_DEC)
void decode_kernel(const float* __restrict__ preds,
                   const float* __restrict__ anchors,
                   float* __restrict__ cboxes,
                   float* __restrict__ cscores,
                   float* __restrict__ ccls,
                   int*   __restrict__ counter)
{
    __shared__ __align__(16) float smem[BPB * VALS];   // 10880 bytes
    const int tid  = threadIdx.x;
    const int lane = tid & 31;
    const int wave = tid >> 5;

    // stage 32 boxes' worth of preds into LDS asynchronously (exactly 680 b128 chunks)
    const char* gbase = (const char*)(preds + (size_t)blockIdx.x * (BPB * VALS));
    unsigned lds_base = (unsigned)(unsigned long long)(void*)smem;   // low 32 bits = LDS offset
#pragma unroll
    for (int k = 0; k < 3; ++k) {
        int c = tid + k * THREADS_DEC;
        if (c < CHUNKS) async_copy16(lds_base + (unsigned)c * 16u, gbase + (size_t)c * 16);
    }
    async_wait0();
    __syncthreads();

    const float NINF = -__builtin_inff();

#pragma unroll
    for (int k = 0; k < BPB / 8; ++k) {
        const int slot = wave + 8 * k;                 // box slot within block
        const float* x = &smem[slot * VALS];
        // lane l covers values l, l+32, l+64 of this box (85 valid)
        float a0 = x[lane];
        float a1 = x[lane + 32];
        float a2 = (lane < 21) ? x[lane + 64] : 0.0f;

        // per-lane best class logit; class index = value_index - 5
        float bv = (lane >= 5) ? a0 : NINF;
        int   bi = lane - 5;
        if (a1 > bv)               { bv = a1; bi = lane + 27; }
        if (lane < 21 && a2 > bv)  { bv = a2; bi = lane + 59; }

        // wave32 butterfly argmax (bv also becomes the max logit on all lanes)
#pragma unroll
        for (int off = 16; off > 0; off >>= 1) {
            float ov = __shfl_xor(bv, off, 32);
            int   oi = __shfl_xor(bi, off, 32);
            if (ov > bv || (ov == bv && oi < bi)) { bv = ov; bi = oi; }
        }

        // sum of exp(logit - max) over the 80 classes
        float e = expf(a1 - bv);
        if (lane >= 5) e += expf(a0 - bv);
        if (lane < 21) e += expf(a2 - bv);
#pragma unroll
        for (int off = 16; off > 0; off >>= 1)
            e += __shfl_xor(e, off, 32);

        if (lane == 0) {
            float conf  = 1.0f / (1.0f + expf(-x[4]));
            float score = conf * (1.0f / e);           // conf * max(softmax)
            if (score >= 0.15f) {                      // SCORE_THRESHOLD
                int b    = blockIdx.x * BPB + slot;    // global box id
                int a    = b % 5;                      // anchor
                int cell = b / 5;
                int gx   = cell % 52;
                int gy   = (cell / 52) % 52;
                float cx = (1.0f / (1.0f + expf(-x[0])) + (float)gx) / GRID_WH;
                float cy = (1.0f / (1.0f + expf(-x[1])) + (float)gy) / GRID_WH;
                float w  = expf(x[2]) * anchors[2 * a + 0] / GRID_WH;
                float h  = expf(x[3]) * anchors[2 * a + 1] / GRID_WH;
                int o = atomicAdd(counter, 1);
                cboxes[4 * o + 0] = cy - h * 0.5f;     // ymin
                cboxes[4 * o + 1] = cx - w * 0.5f;     // xmin
                cboxes[4 * o + 2] = cy + h * 0.5f;     // ymax
                cboxes[4 * o + 3] = cx + w * 0.5f;     // xmax
                cscores[o] = score;
                ccls[o]    = (float)bi;
            }
        }
    }
}

// ---------------- kernel 2: greedy NMS, 10 rounds, single workgroup ----------------
__global__ __launch_bounds__(1024)
void nms_kernel(const float* __restrict__ cboxes,
                float* __restrict__ cscores,
                const float* __restrict__ ccls,
                const int* __restrict__ counter,
                float* __restrict__ out)
{
    __shared__ float s_val[32];
    __shared__ int   s_idx[32];
    __shared__ float s_box[4];
    __shared__ int   s_sel;
    const int tid = threadIdx.x;
    int M = *counter;
    if (M > NBOX) M = NBOX;
    const float NINF = -__builtin_inff();

    for (int r = 0; r < 10; ++r) {
        // ---- block-wide argmax over compacted scores ----
        float bv = NINF; int bi = -1;
        for (int i = tid; i < M; i += 1024) {
            float s = cscores[i];
            if (s > bv) { bv = s; bi = i; }
        }
#pragma unroll
        for (int off = 16; off > 0; off >>= 1) {
            float ov = __shfl_xor(bv, off, 32);
            int   oi = __shfl_xor(bi, off, 32);
            if (ov > bv) { bv = ov; bi = oi; }
        }
        if ((tid & 31) == 0) { s_val[tid >> 5] = bv; s_idx[tid >> 5] = bi; }
        __syncthreads();
        if (tid == 0) {
            float best = NINF; int besti = -1;
            for (int w2 = 0; w2 < 32; ++w2)
                if (s_val[w2] > best) { best = s_val[w2]; besti = s_idx[w2]; }
            bool valid = (besti >= 0) && (best > -5.0e8f);   // s > NEG/2
            float b0 = 0.f, b1 = 0.f, b2 = 0.f, b3 = 0.f;
            if (besti >= 0) {
                b0 = cboxes[4 * besti + 0]; b1 = cboxes[4 * besti + 1];
                b2 = cboxes[4 * besti + 2]; b3 = cboxes[4 * besti + 3];
                cscores[besti] = NEG_F;                      // .at[i].set(NEG)
            }
            if (valid) {
                out[6 * r + 0] = b0; out[6 * r + 1] = b1;
                out[6 * r + 2] = b2; out[6 * r + 3] = b3;
                out[6 * r + 4] = best; out[6 * r + 5] = ccls[besti];
            } else {
                for (int q = 0; q < 6; ++q) out[6 * r + q] = 0.0f;
            }
            s_box[0] = b0; s_box[1] = b1; s_box[2] = b2; s_box[3] = b3;
            s_sel = valid ? besti : -1;   // suppression is a no-op when invalid
        }
        __syncthreads();
        // ---- suppress IoU > 0.5 against the selected box ----
        if (s_sel >= 0) {
            float ymin = s_box[0], xmin = s_box[1], ymax = s_box[2], xmax = s_box[3];
            float area1 = fmaxf(ymax - ymin, 0.0f) * fmaxf(xmax - xmin, 0.0f);
            for (int i = tid; i < M; i += 1024) {
                float y0 = cboxes[4 * i + 0], x0 = cboxes[4 * i + 1];
                float y1 = cboxes[4 * i + 2], x1 = cboxes[4 * i + 3];
                float ity = fmaxf(ymin, y0), itx = fmaxf(xmin, x0);
                float iby = fminf(ymax, y1), ibx = fminf(xmax, x1);
                float inter = fmaxf(iby - ity, 0.0f) * fmaxf(ibx - itx, 0.0f);
                float area2 = fmaxf(y1 - y0, 0.0f) * fmaxf(x1 - x0, 0.0f);
                float iou = inter / (area1 + area2 - inter + 1e-9f);
                if (iou > 0.5f) cscores[i] = NEG_F;
            }
        }
        __syncthreads();
    }
}

// ---------------- launch ----------------
extern "C" void kernel_launch(void* const* d_in, const int* in_sizes, int n_in,
                              void* d_out, int out_size, void* d_ws, size_t ws_size,
                              hipStream_t stream)
{
    const float* preds   = (const float*)d_in[0];   // (16,52,52,425) fp32
    const float* anchors = (const float*)d_in[1];   // (5,2) fp32

    // workspace layout (floats): cboxes[4N] | cscores[N] | ccls[N] | counter(int)
    float* ws      = (float*)d_ws;
    float* cboxes  = ws;
    float* cscores = ws + 4 * (size_t)NBOX;
    float* ccls    = ws + 5 * (size_t)NBOX;
    int*   counter = (int*)(ws + 6 * (size_t)NBOX);

    hipMemsetAsync(counter, 0, sizeof(int), stream);
    decode_kernel<<<NBOX / BPB, THREADS_DEC, 0, stream>>>(preds, anchors,
                                                          cboxes, cscores, ccls, counter);
    nms_kernel<<<1, 1024, 0, stream>>>(cboxes, cscores, ccls, counter, (float*)d_out);
}